// Quan_adder2d_71545565217400
// MI455X (gfx1250) — compile-verified
//
#include <hip/hip_runtime.h>
#include <stdint.h>

// ---------------------------------------------------------------------------
// Quan_adder2d: fake-quant (16-bit global on x, 8-bit per-channel on W) +
// L1-distance ("adder") 3x3 conv, stride 1, pad 1.
//
// Exact restructure used here:
//   |w - x| = w + x - 2*min(w,x)
//   out[n,f,ho,wo] = -( Sw[f] + Sx[n,ho,wo] - 2 * sum_k min(qW[f,k], xpatch[k]) )
// Sw  : per-filter weight sums (prep kernel)
// Sx  : padded patch sums, computed via V_WMMA_F32_16X16X4_F32 (ones x slab)
// min : VALU inner loop (same cost as the abs-diff form)
// ---------------------------------------------------------------------------

#define C_IN      64
#define H_IN      28
#define W_IN      28
#define KDIM      576          // C_IN * 3 * 3
#define FT        16           // filters per LDS tile
#define NTHREADS  256
#define STAT_BLOCKS 128

// workspace layout (floats)
#define WS_SCALE  0
#define WS_ZP     1
#define WS_PMIN   16
#define WS_PMAX   (WS_PMIN + STAT_BLOCKS)
#define WS_QW     (WS_PMAX + STAT_BLOCKS)   // 272 floats -> 1088 B (16B aligned)
// per-filter weight sums live at WS_QW + F*KDIM (F known at runtime)

// x slab in LDS: xs[c*96 + r*32 + (col+1)], col in [-1,28]; +8 floats overrun pad
#define XS_ELEMS  (C_IN * 3 * 32 + 8)

typedef __attribute__((ext_vector_type(2))) float v2f;
typedef __attribute__((ext_vector_type(8))) float v8f;

// ---------------- CDNA5 async-DMA helpers (ASYNCcnt path) ------------------
__device__ __forceinline__ uint32_t lds_addr32(const void* p) {
  // generic pointer into LDS aperture: low 32 bits == wave-relative LDS offset
  return (uint32_t)(uintptr_t)p;
}
__device__ __forceinline__ void async_copy_b32(uint32_t lds_off, const void* gptr) {
  asm volatile("global_load_async_to_lds_b32 %0, %1, off"
               :: "v"(lds_off), "v"(gptr) : "memory");
}
__device__ __forceinline__ void async_copy_b128(uint32_t lds_off, const void* gptr) {
  asm volatile("global_load_async_to_lds_b128 %0, %1, off"
               :: "v"(lds_off), "v"(gptr) : "memory");
}
__device__ __forceinline__ void wait_async_zero() {
  asm volatile("s_wait_asynccnt 0x0" ::: "memory");
}

// --------------------------- pass 1: x min/max -----------------------------
__global__ __launch_bounds__(NTHREADS)
void xstat_kernel(const float* __restrict__ x, int n_elem, float* __restrict__ ws) {
  __shared__ float smn[NTHREADS];
  __shared__ float smx[NTHREADS];
  float mn = 3.4e38f, mx = -3.4e38f;
  for (int i = blockIdx.x * NTHREADS + threadIdx.x; i < n_elem;
       i += gridDim.x * NTHREADS) {
    float v = x[i];
    mn = fminf(mn, v);
    mx = fmaxf(mx, v);
  }
  smn[threadIdx.x] = mn; smx[threadIdx.x] = mx;
  __syncthreads();
  for (int s = NTHREADS / 2; s > 0; s >>= 1) {
    if (threadIdx.x < s) {
      smn[threadIdx.x] = fminf(smn[threadIdx.x], smn[threadIdx.x + s]);
      smx[threadIdx.x] = fmaxf(smx[threadIdx.x], smx[threadIdx.x + s]);
    }
    __syncthreads();
  }
  if (threadIdx.x == 0) {
    ws[WS_PMIN + blockIdx.x] = smn[0];
    ws[WS_PMAX + blockIdx.x] = smx[0];
  }
}

// --- pass 2: block 0 finalizes x scale/zp; blocks 1..F quantize W rows -----
__global__ __launch_bounds__(NTHREADS)
void prep_kernel(const float* __restrict__ Wt, float* __restrict__ ws) {
  __shared__ float smn[NTHREADS];
  __shared__ float smx[NTHREADS];
  const int tid = threadIdx.x;
  const int b = blockIdx.x;
  const int F = (int)gridDim.x - 1;

  float mn = 3.4e38f, mx = -3.4e38f;
  if (b == 0) {
    if (tid < STAT_BLOCKS) { mn = ws[WS_PMIN + tid]; mx = ws[WS_PMAX + tid]; }
  } else {
    const float* wf = Wt + (size_t)(b - 1) * KDIM;
    for (int i = tid; i < KDIM; i += NTHREADS) {
      float w = wf[i];
      mn = fminf(mn, w);
      mx = fmaxf(mx, w);
    }
  }
  smn[tid] = mn; smx[tid] = mx;
  __syncthreads();
  for (int s = NTHREADS / 2; s > 0; s >>= 1) {
    if (tid < s) {
      smn[tid] = fminf(smn[tid], smn[tid + s]);
      smx[tid] = fmaxf(smx[tid], smx[tid + s]);
    }
    __syncthreads();
  }

  if (b == 0) {
    if (tid == 0) {
      float scale = fmaxf((smx[0] - smn[0]) * (1.0f / 65535.0f), 1e-12f);
      float zp = rintf(-smn[0] / scale);
      ws[WS_SCALE] = scale;
      ws[WS_ZP]    = zp;
    }
  } else {
    const int f = b - 1;
    const float* wf = Wt + (size_t)f * KDIM;
    float scale = fmaxf((smx[0] - smn[0]) * (1.0f / 255.0f), 1e-12f);
    float zp = rintf(-smn[0] / scale);
    float ssum = 0.0f;
    for (int i = tid; i < KDIM; i += NTHREADS) {
      float w = wf[i];
      float q = fminf(fmaxf(rintf(w / scale) + zp, 0.0f), 255.0f);
      float dq = (q - zp) * scale;
      ws[WS_QW + (size_t)f * KDIM + i] = dq;
      ssum += dq;
    }
    // per-filter weight sum Sw[f]
    smn[tid] = ssum;
    __syncthreads();
    for (int s = NTHREADS / 2; s > 0; s >>= 1) {
      if (tid < s) smn[tid] = smn[tid] + smn[tid + s];
      __syncthreads();
    }
    if (tid == 0) ws[WS_QW + (size_t)F * KDIM + f] = smn[0];
  }
}

// ------------- pass 3: fused quantize-x + L1-distance conv -----------------
__global__ __launch_bounds__(NTHREADS)
void adder_main_kernel(const float* __restrict__ x, const float* __restrict__ ws,
                       float* __restrict__ out, int F) {
  __shared__ __align__(16) float xs[XS_ELEMS];       // 3-row x slab, 64 ch
  __shared__ __align__(16) float wl[FT * KDIM];      // 16-filter weight tile
  __shared__ float colsum[40];                       // WMMA-computed column sums

  const int tid = threadIdx.x;
  const int n  = blockIdx.x / H_IN;
  const int ho = blockIdx.x % H_IN;

  // zero LDS (padding cells must be exactly 0.0f, matching pad-after-quant)
  for (int i = tid; i < XS_ELEMS; i += NTHREADS) xs[i] = 0.0f;
  for (int i = tid; i < 40; i += NTHREADS) colsum[i] = 0.0f;

  // async-DMA the three input rows (all channels) into LDS interior
  const int NXLD = C_IN * 3 * W_IN;                  // 5376 scalar transfers
  for (int t = tid; t < NXLD; t += NTHREADS) {
    int c   = t / (3 * W_IN);
    int rr  = (t / W_IN) % 3;
    int col = t % W_IN;
    int row = ho - 1 + rr;
    if (row >= 0 && row < H_IN) {
      const float* gp = x + ((((size_t)n * C_IN + c) * H_IN + row) * W_IN + col);
      async_copy_b32(lds_addr32(&xs[c * 96 + rr * 32 + col + 1]), gp);
    }
  }
  // async-DMA first weight tile (contiguous, 16B aligned -> b128)
  const float* qw = ws + WS_QW;
  for (int t = tid; t < (FT * KDIM) / 4; t += NTHREADS) {
    async_copy_b128(lds_addr32(&wl[t * 4]), qw + (size_t)t * 4);
  }
  wait_async_zero();
  __syncthreads();

  // in-place 16-bit fake-quant of the loaded interior (pads stay 0.0)
  const float sx = ws[WS_SCALE];
  const float zx = ws[WS_ZP];
  const float inv_sx = 1.0f / sx;
  for (int t = tid; t < NXLD; t += NTHREADS) {
    int c   = t / (3 * W_IN);
    int rr  = (t / W_IN) % 3;
    int col = t % W_IN;
    int row = ho - 1 + rr;
    if (row >= 0 && row < H_IN) {
      int idx = c * 96 + rr * 32 + col + 1;
      float v = xs[idx];
      float q = fminf(fmaxf(rintf(v * inv_sx) + zx, 0.0f), 65535.0f);
      xs[idx] = (q - zx) * sx;
    }
  }
  __syncthreads();

  // ---- WMMA: column sums of the 192x32 slab via ones(16x4) x B(4x16) ----
  // colsum[m] = sum over r<192 of xs[(r/3)*96 + (r%3)*32 + m]
  // wave 0 -> columns 0..15, wave 1 -> columns 16..31 (tail cols are zero pad)
  {
    const int wave = tid >> 5;
    const int lane = tid & 31;
    if (wave < 2) {
      v2f a; a.x = 1.0f; a.y = 1.0f;          // A = all-ones 16x4
      v8f c = {};
      const int nn  = lane & 15;              // column within group
      const int kk  = (lane >> 4) << 1;       // B rows {0,1} or {2,3}
      const int col = (wave << 4) + nn;
#pragma unroll 4
      for (int s = 0; s < 48; ++s) {
        int r0 = 4 * s + kk;
        int r1 = r0 + 1;
        v2f b;
        b.x = xs[(r0 / 3) * 96 + (r0 % 3) * 32 + col];
        b.y = xs[(r1 / 3) * 96 + (r1 % 3) * 32 + col];
        c = __builtin_amdgcn_wmma_f32_16x16x4_f32(
                /*neg_a=*/false, a, /*neg_b=*/false, b,
                /*c_mod=*/(short)0, c, /*reuse_a=*/false, /*reuse_b=*/false);
      }
      if (lane < 16) colsum[col] = c[0];      // D[M=0][N=lane] = column sum
    }
  }
  __syncthreads();

  // per-thread output mapping: filter fl = tid%16, pixels wo0 = tid/16, wo0+16
  const int fl  = tid & (FT - 1);
  const int wo0 = tid >> 4;          // 0..15
  const int wo1 = wo0 + 16;          // 16..31 (valid if < 28)
  const int nft = F / FT;
  const float* swv = ws + WS_QW + (size_t)F * KDIM;  // Sw[f]

  // padded patch sums Sx (padded col index == wo .. wo+2)
  const float s2_0 = colsum[wo0] + colsum[wo0 + 1] + colsum[wo0 + 2];
  const float s2_1 = colsum[wo1] + colsum[wo1 + 1] + colsum[wo1 + 2];

  for (int ft = 0; ft < nft; ++ft) {
    if (ft > 0) {
      __syncthreads();               // previous tile fully consumed
      for (int t = tid; t < (FT * KDIM) / 4; t += NTHREADS) {
        async_copy_b128(lds_addr32(&wl[t * 4]),
                        qw + (size_t)ft * FT * KDIM + (size_t)t * 4);
      }
      wait_async_zero();
      __syncthreads();
    }

    float acc0 = 0.0f, acc1 = 0.0f;  // sum of min(w, x)
    const float* wrow = &wl[fl * KDIM];
    for (int c = 0; c < C_IN; ++c) {
      const float* xr = &xs[c * 96];
      const float* wr = &wrow[c * 9];
#pragma unroll
      for (int i = 0; i < 3; ++i) {
#pragma unroll
        for (int j = 0; j < 3; ++j) {
          float w = wr[i * 3 + j];
          acc0 += fminf(w, xr[i * 32 + wo0 + j]);
          acc1 += fminf(w, xr[i * 32 + wo1 + j]);  // overreads land in zero pad
        }
      }
    }
    const int f = ft * FT + fl;
    const float swf = swv[f];
    float* orow = out + (((size_t)n * F + f) * H_IN + ho) * W_IN;
    orow[wo0] = -(swf + s2_0 - 2.0f * acc0);
    if (wo1 < W_IN) orow[wo1] = -(swf + s2_1 - 2.0f * acc1);
  }
}

// ---------------------------------------------------------------------------
extern "C" void kernel_launch(void* const* d_in, const int* in_sizes, int n_in,
                              void* d_out, int out_size, void* d_ws, size_t ws_size,
                              hipStream_t stream) {
  const float* x  = (const float*)d_in[0];
  const float* Wt = (const float*)d_in[1];
  float* out = (float*)d_out;
  float* ws  = (float*)d_ws;

  const int n_x = in_sizes[0];
  const int N   = n_x / (C_IN * H_IN * W_IN);        // 4
  const int F   = in_sizes[1] / KDIM;                // 64

  xstat_kernel<<<STAT_BLOCKS, NTHREADS, 0, stream>>>(x, n_x, ws);
  prep_kernel<<<F + 1, NTHREADS, 0, stream>>>(Wt, ws);
  adder_main_kernel<<<N * H_IN, NTHREADS, 0, stream>>>(x, ws, out, F);
}